// TreeMapping_67671504715946
// MI455X (gfx1250) — compile-verified
//
#include <hip/hip_runtime.h>
#include <math.h>

// TreeMapping on MI455X (gfx1250).
// path_prob[b,j] = prod_k comb[b,j,k] with comb in {x, 1-x, 1}
//   == exp( log(x[b,:]) . R[j,:] + log(1-x[b,:]) . L[j,:] )
// -> two fused fp32 GEMMs (M=512, N=512 padded, K=256 padded) on
//    V_WMMA_F32_16X16X4_F32, full fp32 accumulation.

typedef __attribute__((ext_vector_type(2))) float v2f;
typedef __attribute__((ext_vector_type(8))) float v8f;

#define BATCH 512
#define NN    255   // N_NODES
#define NL    256   // N_LEAVES
#define NROWS 511   // combined matrix rows (leaves + nodes)
#define KP    256   // padded K
#define NT    32    // 512/16 tiles in N
#define MT    32    // 512/16 tiles in M

// Kernel 1: build padded log-domain A matrices and padded 0/1 B matrices.
__global__ __launch_bounds__(256) void tree_prep(
    const float* __restrict__ x,
    const float* __restrict__ rbin,
    const float* __restrict__ lbin,
    float* __restrict__ Lx, float* __restrict__ L1,
    float* __restrict__ Rp, float* __restrict__ Lp)
{
    int idx = blockIdx.x * 256 + threadIdx.x;   // 512*256 threads
    int c = idx & (KP - 1);
    int r = idx >> 8;

    float lx = 0.0f, l1 = 0.0f;
    if (c < NN) {
        float v = x[r * NN + c];
        // clamp so a (theoretical) log(0) = -inf cannot make -inf*0 = NaN in the GEMM
        lx = fmaxf(logf(v),        -1.0e30f);
        l1 = fmaxf(logf(1.0f - v), -1.0e30f);
    }
    Lx[idx] = lx;
    L1[idx] = l1;

    float rv = 0.0f, lv = 0.0f;
    if (r < NROWS && c < NN) {
        rv = rbin[r * NN + c];
        lv = lbin[r * NN + c];
    }
    Rp[idx] = rv;
    Lp[idx] = lv;
}

// Kernel 2: one wave per 16x16 output tile; K loop in steps of 4 via
// v_wmma_f32_16x16x4_f32 (ISA 7.12.2 layouts):
//   A 16x4 f32:  lane = (l&15) is M row; VGPR0 = K=(l>>4)*2, VGPR1 = K+1
//   B 4x16 f32:  lane = (l&15) is N col; VGPR0 = K=(l>>4)*2, VGPR1 = K+1
//   C/D 16x16:   VGPR v -> M = v + 8*(l>=16), N = l&15
__global__ __launch_bounds__(256) void tree_wmma(
    const float* __restrict__ Lx, const float* __restrict__ L1,
    const float* __restrict__ Rp, const float* __restrict__ Lp,
    float* __restrict__ out)
{
    const int lane = threadIdx.x & 31;
    const int wave = threadIdx.x >> 5;
    const int tile = blockIdx.x * 8 + wave;     // 0..1023
    const int nt = tile & (NT - 1);
    const int mt = tile >> 5;
    const int m0 = mt * 16;
    const int n0 = nt * 16;

    const int rc = lane & 15;                   // A row / B col
    const int kk = (lane >> 4) << 1;            // 0 or 2

    const float* ap  = Lx + (size_t)(m0 + rc) * KP + kk;
    const float* a2p = L1 + (size_t)(m0 + rc) * KP + kk;
    const float* bp  = Rp + (size_t)(n0 + rc) * KP + kk;
    const float* b2p = Lp + (size_t)(n0 + rc) * KP + kk;

    v8f acc = {};
#pragma unroll 8
    for (int k0 = 0; k0 < KP; k0 += 4) {
        v2f a  = *(const v2f*)(ap  + k0);
        v2f a2 = *(const v2f*)(a2p + k0);
        v2f br = *(const v2f*)(bp  + k0);
        v2f bl = *(const v2f*)(b2p + k0);
        acc = __builtin_amdgcn_wmma_f32_16x16x4_f32(
                  false, a,  false, br, (short)0, acc, false, false);
        acc = __builtin_amdgcn_wmma_f32_16x16x4_f32(
                  false, a2, false, bl, (short)0, acc, false, false);
    }

    // Epilogue: exp() and scatter to [leaf | node] halves of d_out.
    const int j     = n0 + rc;                  // combined-matrix row (< 512)
    const int mbase = m0 + ((lane >> 4) << 3);
#pragma unroll
    for (int v = 0; v < 8; ++v) {
        int m = mbase + v;
        float e = expf(acc[v]);
        if (j < NROWS) {
            if (j < NL) out[(size_t)m * NL + j] = e;                       // leaf_probs
            else        out[(size_t)BATCH * NL + (size_t)m * NN + (j - NL)] = e; // node_probs
        }
    }
}

extern "C" void kernel_launch(void* const* d_in, const int* in_sizes, int n_in,
                              void* d_out, int out_size, void* d_ws, size_t ws_size,
                              hipStream_t stream) {
    const float* x    = (const float*)d_in[0];   // 512 x 255
    const float* rbin = (const float*)d_in[1];   // 511 x 255
    const float* lbin = (const float*)d_in[2];   // 511 x 255
    // d_in[3] (no_bin) contributes log(1)=0 terms -> unused.

    float* ws = (float*)d_ws;                    // needs 4 * 512*256 * 4B = 2 MB
    float* Lx = ws;
    float* L1 = ws + 512 * 256;
    float* Rp = ws + 2 * 512 * 256;
    float* Lp = ws + 3 * 512 * 256;

    tree_prep<<<512, 256, 0, stream>>>(x, rbin, lbin, Lx, L1, Rp, Lp);
    tree_wmma<<<128, 256, 0, stream>>>(Lx, L1, Rp, Lp, (float*)d_out);
}